// VectorQuantizer_65352222376129
// MI455X (gfx1250) — compile-verified
//
#include <hip/hip_runtime.h>
#include <math.h>
#include <stdint.h>

typedef float v2f __attribute__((ext_vector_type(2)));
typedef float v4f __attribute__((ext_vector_type(4)));
typedef float v8f __attribute__((ext_vector_type(8)));
typedef int   v4i __attribute__((ext_vector_type(4)));

#define DIMD 256
#define NUMK 8192
#define NUMN 32768
#define WAVES 8
#define ROWS_PER_WG (WAVES * 16)   // 128

// ---- CDNA5 async global->LDS copy (bypasses VGPRs, tracked by ASYNCcnt) ----
__device__ __forceinline__ void async_copy_b128(const float* gsrc, const float* ldst) {
    // dst = LDS byte offset (low 32 bits of generic pointer), src = 64-bit VA
    asm volatile("global_load_async_to_lds_b128 %0, %1, off"
                 :: "v"((unsigned)(uintptr_t)ldst), "v"(gsrc)
                 : "memory");
}
__device__ __forceinline__ void wait_async0() {
    asm volatile("s_wait_asynccnt 0x0" ::: "memory");
}

// ---------------- zero encodings region + histogram ----------------
__global__ __launch_bounds__(256) void vq_zero_kernel(v4f* enc4, long n4, float* tail2, v4i* hist4) {
    long gid = (long)blockIdx.x * blockDim.x + threadIdx.x;
    long stride = (long)gridDim.x * blockDim.x;
    v4f z = {0.f, 0.f, 0.f, 0.f};
    for (long i = gid; i < n4; i += stride)
        __builtin_nontemporal_store(z, &enc4[i]);
    if (gid < NUMK / 4) {
        v4i zi = {0, 0, 0, 0};
        hist4[gid] = zi;
    }
    if (gid == 0) { tail2[0] = 0.f; tail2[1] = 0.f; }
}

// ---------------- transpose b c h w -> (b h w) c ----------------
__global__ __launch_bounds__(256) void vq_transpose_kernel(const float* __restrict__ in,
                                                           float* __restrict__ flat) {
    __shared__ float tile[32 * 257];  // [w][c], padded to kill bank conflicts
    const int b = blockIdx.x, h = blockIdx.y;
    const int lane = threadIdx.x & 31, g = threadIdx.x >> 5;
    #pragma unroll 4
    for (int cc = 0; cc < 32; ++cc) {
        int c = g * 32 + cc;
        // in[((b*256+c)*32+h)*32 + w]: lanes read contiguous 128B along w
        tile[lane * 257 + c] = in[(((long)b * 256 + c) * 32 + h) * 32 + lane];
    }
    __syncthreads();
    const long n0 = (long)b * 1024 + h * 32;
    #pragma unroll 4
    for (int w = 0; w < 32; ++w)
        flat[(n0 + w) * DIMD + threadIdx.x] = tile[w * 257 + threadIdx.x];
}

// ---------------- codebook row norms ----------------
__global__ __launch_bounds__(256) void vq_enorm_kernel(const float* __restrict__ cb,
                                                       float* __restrict__ enorm) {
    const int wave = threadIdx.x >> 5, lane = threadIdx.x & 31;
    const int k = blockIdx.x * 8 + wave;
    const float* row = cb + (long)k * DIMD;
    float s = 0.f;
    #pragma unroll
    for (int i = 0; i < 8; ++i) { float v = row[lane + 32 * i]; s += v * v; }
    #pragma unroll
    for (int off = 16; off >= 1; off >>= 1) s += __shfl_xor(s, off, 32);
    if (lane == 0) enorm[k] = s;
}

// ---------------- WMMA GEMM + fused argmin ----------------
// A rows resident in VGPRs (128 regs/lane); B tiles double-buffered in LDS via
// CDNA5 async global->LDS copies; one barrier per 16-column tile.
__global__ __launch_bounds__(256) void vq_argmin_kernel(const float* __restrict__ flat,
                                                        const float* __restrict__ cb,
                                                        const float* __restrict__ enorm,
                                                        int* __restrict__ idx_out) {
    // 128 KB: used first to stage 128 rows of x, then first 32 KB reused as the
    // B double-buffer (2 x 16 codebook rows x 256 f32).
    __shared__ float smem[ROWS_PER_WG * DIMD];
    const int t = threadIdx.x;
    const int wave = t >> 5;
    const int lane = t & 31;
    const int l16 = lane & 15;   // M (A) / N (B) index within fragment
    const int hi  = lane >> 4;   // selects K-pair {0,1} vs {2,3}
    const long rowBase = (long)blockIdx.x * ROWS_PER_WG;

    // ---- stage A (async): 128x256 f32 = 8192 x b128, 256 threads x 32 ----
    {
        const float* gA = flat + rowBase * DIMD;
        #pragma unroll
        for (int i = 0; i < 32; ++i)
            async_copy_b128(gA + (t + i * 256) * 4, &smem[(t + i * 256) * 4]);
        wait_async0();
    }
    __syncthreads();

    // ---- hoist this wave's A fragments into registers: 64 x v2f ----
    v2f areg[64];
    {
        const float* aBase = smem + (wave * 16 + l16) * DIMD + 2 * hi;
        #pragma unroll
        for (int i = 0; i < 64; ++i)
            areg[i] = *(const v2f*)(aBase + 4 * i);
    }
    __syncthreads();   // all waves done reading A before B aliases the region

    float best[8];
    int   bidx[8];
    #pragma unroll
    for (int r = 0; r < 8; ++r) { best[r] = 3.4e38f; bidx[r] = 0; }

    // ---- prologue: async-stage B tile 0 into buffer 0 ----
    {
        const float* g0 = cb;
        #pragma unroll
        for (int i = 0; i < 4; ++i)
            async_copy_b128(g0 + (t + i * 256) * 4, &smem[(t + i * 256) * 4]);
        wait_async0();
    }
    __syncthreads();

    for (int kt = 0; kt < NUMK / 16; ++kt) {
        // issue async stage of tile kt+1 into the other buffer
        if (kt + 1 < NUMK / 16) {
            const float* g0 = cb + (long)(kt + 1) * 16 * DIMD;
            const int buf = (kt + 1) & 1;
            #pragma unroll
            for (int i = 0; i < 4; ++i)
                async_copy_b128(g0 + (t + i * 256) * 4,
                                &smem[buf * (16 * DIMD) + (t + i * 256) * 4]);
        }

        // compute on buffer kt&1
        const float* bBase = smem + (kt & 1) * (16 * DIMD) + l16 * DIMD + 2 * hi;
        v8f acc = {};
        #pragma unroll
        for (int i = 0; i < 64; ++i) {
            v2f b = *(const v2f*)(bBase + 4 * i);
            acc = __builtin_amdgcn_wmma_f32_16x16x4_f32(
                false, areg[i], false, b, (short)0, acc, false, false);
        }

        const int kcol = kt * 16 + l16;
        const float en = enorm[kcol];
        #pragma unroll
        for (int r = 0; r < 8; ++r) {
            float score = en - 2.0f * acc[r];
            if (score < best[r]) { best[r] = score; bidx[r] = kcol; }
        }

        wait_async0();     // next tile fully landed in LDS
        __syncthreads();   // everyone done with buffer kt&1 / sees tile kt+1
    }

    // C layout: lanes 0-15 hold rows 0-7 (vgpr r -> M=r), lanes 16-31 rows 8-15.
    // Argmin across the 16 lanes of each half (columns), tie-break lowest index.
    #pragma unroll
    for (int r = 0; r < 8; ++r) {
        #pragma unroll
        for (int off = 8; off >= 1; off >>= 1) {
            float ob = __shfl_xor(best[r], off, 32);
            int   oi = __shfl_xor(bidx[r], off, 32);
            if (ob < best[r] || (ob == best[r] && oi < bidx[r])) { best[r] = ob; bidx[r] = oi; }
        }
    }
    if (l16 == 0) {
        const int m0 = wave * 16 + hi * 8;
        #pragma unroll
        for (int r = 0; r < 8; ++r)
            idx_out[rowBase + m0 + r] = bidx[r];
    }
}

// ---------------- gather quantized, one-hot scatter, MSE partials ----------------
__global__ __launch_bounds__(256) void vq_gather_kernel(const float* __restrict__ flat,
                                                        const float* __restrict__ cb,
                                                        const int* __restrict__ idx,
                                                        float* __restrict__ out_q,
                                                        float* __restrict__ out_enc,
                                                        int* __restrict__ hist,
                                                        float* __restrict__ partials) {
    __shared__ float wsum[WAVES];
    const int wave = threadIdx.x >> 5, lane = threadIdx.x & 31;
    const long n = (long)blockIdx.x * 8 + wave;
    const int k = idx[n];
    const float* q = cb + (long)k * DIMD;
    const float* x = flat + n * DIMD;
    float acc = 0.f;
    #pragma unroll
    for (int i = 0; i < 8; ++i) {
        int c = lane + 32 * i;
        float qv = q[c], xv = x[c];
        out_q[n * DIMD + c] = qv;
        float d = qv - xv;
        acc += d * d;
    }
    #pragma unroll
    for (int off = 16; off >= 1; off >>= 1) acc += __shfl_xor(acc, off, 32);
    if (lane == 0) {
        wsum[wave] = acc;
        atomicAdd(&hist[k], 1);
        __builtin_nontemporal_store(1.0f, &out_enc[n * (long)NUMK + k]);
    }
    __syncthreads();
    if (threadIdx.x == 0) {
        float s = 0.f;
        #pragma unroll
        for (int w = 0; w < WAVES; ++w) s += wsum[w];
        partials[blockIdx.x] = s;  // fixed-order, deterministic
    }
}

// ---------------- finalize loss + perplexity ----------------
__global__ __launch_bounds__(256) void vq_finalize_kernel(const float* __restrict__ partials,
                                                          const int* __restrict__ hist,
                                                          float* __restrict__ d_out) {
    __shared__ float red[256];
    const int t = threadIdx.x;
    float s = 0.f;
    #pragma unroll
    for (int i = 0; i < 16; ++i) s += partials[t + 256 * i];
    red[t] = s;
    __syncthreads();
    for (int off = 128; off >= 1; off >>= 1) {
        if (t < off) red[t] += red[t + off];
        __syncthreads();
    }
    if (t == 0) d_out[0] = 1.25f * red[0] / (float)((double)NUMN * DIMD);
    __syncthreads();
    float e = 0.f;
    #pragma unroll
    for (int i = 0; i < 32; ++i) {
        float p = (float)hist[t + 256 * i] * (1.0f / (float)NUMN);
        e += p * logf(p + 1e-10f);
    }
    red[t] = e;
    __syncthreads();
    for (int off = 128; off >= 1; off >>= 1) {
        if (t < off) red[t] += red[t + off];
        __syncthreads();
    }
    if (t == 0) d_out[1 + (long)NUMN * DIMD] = expf(-red[0]);
}

extern "C" void kernel_launch(void* const* d_in, const int* in_sizes, int n_in,
                              void* d_out, int out_size, void* d_ws, size_t ws_size,
                              hipStream_t stream) {
    const float* inputs   = (const float*)d_in[0];   // [32,256,32,32]
    const float* codebook = (const float*)d_in[1];   // [8192,256]
    float* out = (float*)d_out;

    // d_out layout: [0]=loss, [1..8388608]=quantized, [8388609]=perplexity,
    //               [8388610..)=encodings (N*K one-hot)
    const long ND = (long)NUMN * DIMD;               // 8388608
    float* out_q   = out + 1;
    float* out_enc = out + ND + 2;

    // workspace layout (floats)
    float* ws       = (float*)d_ws;
    float* flat     = ws;                // 8388608
    float* enorm    = ws + 8388608;      // 8192
    float* partials = ws + 8396800;      // 4096
    int*   idx      = (int*)(ws + 8400896);  // 32768
    int*   hist     = (int*)(ws + 8433664);  // 8192

    // 1) zero encodings region [ND, ND+2+N*K) and histogram
    v4f* enc4 = (v4f*)(out + ND);                       // 16B-aligned
    long n4 = ((long)NUMN * NUMK) / 4;                  // 67,108,864 v4f
    float* tail2 = out + ND + (long)NUMN * NUMK;        // last 2 floats of region
    vq_zero_kernel<<<2048, 256, 0, stream>>>(enc4, n4, tail2, (v4i*)hist);

    // 2) transpose to [N, D]
    vq_transpose_kernel<<<dim3(32, 32), 256, 0, stream>>>(inputs, flat);

    // 3) codebook norms
    vq_enorm_kernel<<<NUMK / 8, 256, 0, stream>>>(codebook, enorm);

    // 4) WMMA GEMM + argmin
    vq_argmin_kernel<<<NUMN / ROWS_PER_WG, 256, 0, stream>>>(flat, codebook, enorm, idx);

    // 5) gather / one-hot / MSE partials
    vq_gather_kernel<<<NUMN / 8, 256, 0, stream>>>(flat, codebook, idx, out_q, out_enc,
                                                   hist, partials);

    // 6) loss + perplexity
    vq_finalize_kernel<<<1, 256, 0, stream>>>(partials, hist, out);
}